// PCloss_59356448030908
// MI455X (gfx1250) — compile-verified
//
#include <hip/hip_runtime.h>
#include <math.h>

#define PI_F 3.14159265358979323846f
constexpr int NPIX = 1 << 20;   // 1024*1024

typedef float v2f  __attribute__((ext_vector_type(2)));
typedef float v8f  __attribute__((ext_vector_type(8)));
typedef unsigned int u32x4 __attribute__((ext_vector_type(4)));
typedef int   i32x4 __attribute__((ext_vector_type(4)));
typedef int   i32x8 __attribute__((ext_vector_type(8)));

#if defined(__gfx1250__) && __has_builtin(__builtin_amdgcn_tensor_load_to_lds)
#define USE_TDM 1
#else
#define USE_TDM 0
#endif

// ---------------------------------------------------------------------------
// CDNA5 f32 WMMA: D(16x16) = A(16x4) * B(4x16) + C   (wave32, all lanes active)
// ---------------------------------------------------------------------------
static __device__ __forceinline__ v8f wmma4(v2f a, v2f b, v8f c) {
  return __builtin_amdgcn_wmma_f32_16x16x4_f32(
      /*neg_a=*/false, a, /*neg_b=*/false, b,
      /*c_mod=*/(short)0, c, /*reuse_a=*/false, /*reuse_b=*/false);
}

#if USE_TDM
// Tensor-DMA descriptor for one strided column of 1024 float2 elements:
// data_size=8B, tensor 1024x1024, tile 1 x 1024, dim0 stride = 1024 elements.
static __device__ __forceinline__ void tdm_desc(unsigned long long ga, unsigned lds,
                                                u32x4& g0, i32x8& g1) {
  g0.x = 1u;                                   // count=1, user descriptor
  g0.y = lds;                                  // LDS byte address
  g0.z = (unsigned)ga;                         // global_addr[31:0]
  g0.w = (unsigned)(ga >> 32) | 0x80000000u;   // global_addr[56:32] | type=2
  g1[0] = 0x00030000;                          // data_size=3 (8B)
  g1[1] = (int)(1024u << 16);                  // tensor_dim0[15:0] in [63:48]
  g1[2] = (int)(1024u << 16);                  // tensor_dim1[15:0] in [95:80]
  g1[3] = (int)(1u << 16);                     // tile_dim0=1 in [127:112]
  g1[4] = 1024;                                // tile_dim1=1024 in [143:128]
  g1[5] = 1024;                                // tensor_dim0_stride[31:0]
  g1[6] = 0;
  g1[7] = 0;
}

static __device__ __forceinline__ void tdm_load(u32x4 g0, i32x8 g1) {
  i32x4 z4 = {0, 0, 0, 0};
#if __clang_major__ >= 23
  i32x8 z8 = {0, 0, 0, 0, 0, 0, 0, 0};
  __builtin_amdgcn_tensor_load_to_lds(g0, g1, z4, z4, z8, 0);
#else
  __builtin_amdgcn_tensor_load_to_lds(g0, g1, z4, z4, 0);
#endif
}

static __device__ __forceinline__ void tdm_store(u32x4 g0, i32x8 g1) {
  i32x4 z4 = {0, 0, 0, 0};
#if __clang_major__ >= 23
  i32x8 z8 = {0, 0, 0, 0, 0, 0, 0, 0};
  __builtin_amdgcn_tensor_store_from_lds(g0, g1, z4, z4, z8, 0);
#else
  __builtin_amdgcn_tensor_store_from_lds(g0, g1, z4, z4, 0);
#endif
}
#endif // USE_TDM

#define FLAG_REAL_IN   1
#define FLAG_CONJ_IN   2
#define FLAG_CONJ_OUT  4
#define FLAG_SPLIT_OUT 8

// ---------------------------------------------------------------------------
// 1024-point FFT (four-step, 1024 = 32x32) per block.  8 waves: each wave owns
// one 16x16 output tile of one part (re/im) of the 32x32 complex matmul.
// estride==1: coalesced row pass.  estride==1024: column pass, the strided
// column gather/scatter is delegated to the Tensor Data Mover (TENSORcnt).
// ---------------------------------------------------------------------------
__global__ __launch_bounds__(256)
void fft1024_kernel(const float* __restrict__ inR,
                    const float2* __restrict__ inC,
                    float2* __restrict__ outC,
                    float* __restrict__ outO,   // real part plane  (O = Re z)
                    float* __restrict__ outE,   // imag part plane  (E = Im z)
                    int estride, int vstride, int flags) {
  __shared__ float wr_[32][32], wi_[32][32];   // DFT32 matrix (symmetric)
  __shared__ float xr_[32][33], xi_[32][33];   // input, reused for output Z
  __shared__ float yr_[32][33], yi_[32][33];   // stage-A result
  __shared__ float2 cbuf[1024];                // TDM staging (column pass)

  const int tid = threadIdx.x;
  const size_t base = (size_t)blockIdx.x * (size_t)vstride;

  // DFT32 table: W[k][n] = exp(-2*pi*i*k*n/32)
  #pragma unroll
  for (int j = 0; j < 4; ++j) {
    int e = tid * 4 + j, k = e >> 5, n = e & 31;
    float s, c, ang = -2.0f * PI_F * (float)(k * n) * (1.0f / 32.0f);
    __sincosf(ang, &s, &c);
    wr_[k][n] = c; wi_[k][n] = s;
  }

  // ---- load x[n1*32+n2] -> Xmat[n1][n2] ----
  if (estride == 1) {
    #pragma unroll
    for (int j = 0; j < 4; ++j) {
      int e = tid * 4 + j, n1 = e >> 5, n2 = e & 31;
      float re, im;
      if (flags & FLAG_REAL_IN) {
        re = inR[base + (size_t)e]; im = 0.0f;
      } else {
        float2 v = inC[base + (size_t)e];
        re = v.x; im = (flags & FLAG_CONJ_IN) ? -v.y : v.y;
      }
      xr_[n1][n2] = re; xi_[n1][n2] = im;
    }
  } else {
#if USE_TDM
    if (tid < 32) {                       // wave 0 issues the DMA gather
      u32x4 g0; i32x8 g1;
      tdm_desc((unsigned long long)(const void*)(inC + base),
               (unsigned)(size_t)(void*)cbuf, g0, g1);
      tdm_load(g0, g1);
    }
    __builtin_amdgcn_s_wait_tensorcnt(0);
    __syncthreads();
    #pragma unroll
    for (int j = 0; j < 4; ++j) {
      int e = tid * 4 + j, n1 = e >> 5, n2 = e & 31;
      float2 v = cbuf[e];
      xr_[n1][n2] = v.x;
      xi_[n1][n2] = (flags & FLAG_CONJ_IN) ? -v.y : v.y;
    }
#else
    #pragma unroll
    for (int j = 0; j < 4; ++j) {
      int e = tid * 4 + j, n1 = e >> 5, n2 = e & 31;
      float2 v = inC[base + (size_t)e * estride];
      xr_[n1][n2] = v.x;
      xi_[n1][n2] = (flags & FLAG_CONJ_IN) ? -v.y : v.y;
    }
#endif
  }
  __syncthreads();

  const int lane = tid & 31;
  const int wave = tid >> 5;
  const int part = wave >> 2;          // 0 = real result, 1 = imag result
  const int ti   = (wave >> 1) & 1;    // output row tile
  const int tj   = wave & 1;           // output col tile
  const int M    = lane & 15;          // A row / D col within tile
  const int kh   = (lane >> 4) << 1;   // K-pair base within chunk
  const float s2 = part ? 1.0f : -1.0f;

  // ---- stage A:  Y = W32 * X  (Yr = Wr Xr - Wi Xi ; Yi = Wr Xi + Wi Xr) ----
  {
    const float (*B1)[33] = part ? xi_ : xr_;
    const float (*B2)[33] = part ? xr_ : xi_;
    v8f acc = {0.f, 0.f, 0.f, 0.f, 0.f, 0.f, 0.f, 0.f};
    #pragma unroll
    for (int kk = 0; kk < 8; ++kk) {
      int k0 = kk * 4 + kh;
      v2f a1, a2, b1, b2;
      a1.x = wr_[ti * 16 + M][k0];        a1.y = wr_[ti * 16 + M][k0 + 1];
      a2.x = s2 * wi_[ti * 16 + M][k0];   a2.y = s2 * wi_[ti * 16 + M][k0 + 1];
      b1.x = B1[k0][tj * 16 + M];         b1.y = B1[k0 + 1][tj * 16 + M];
      b2.x = B2[k0][tj * 16 + M];         b2.y = B2[k0 + 1][tj * 16 + M];
      acc = wmma4(a1, b1, acc);
      acc = wmma4(a2, b2, acc);
    }
    float (*D)[33] = part ? yi_ : yr_;
    #pragma unroll
    for (int v = 0; v < 8; ++v)
      D[ti * 16 + v + (lane >> 4) * 8][tj * 16 + M] = acc[v];
  }
  __syncthreads();

  // ---- twiddle: Y[k1][n2] *= exp(-2*pi*i*k1*n2/1024) ----
  #pragma unroll
  for (int j = 0; j < 4; ++j) {
    int e = tid * 4 + j, k1 = e >> 5, n2 = e & 31;
    float s, c, ang = -2.0f * PI_F * (float)(k1 * n2) * (1.0f / 1024.0f);
    __sincosf(ang, &s, &c);
    float ar = yr_[k1][n2], ai = yi_[k1][n2];
    yr_[k1][n2] = ar * c - ai * s;
    yi_[k1][n2] = ar * s + ai * c;
  }
  __syncthreads();

  // ---- stage B:  Z = Y * W32  (reuse xr_/xi_ as Z) ----
  {
    const float (*Bm1)[32] = part ? wi_ : wr_;
    const float (*Bm2)[32] = part ? wr_ : wi_;
    v8f acc = {0.f, 0.f, 0.f, 0.f, 0.f, 0.f, 0.f, 0.f};
    #pragma unroll
    for (int kk = 0; kk < 8; ++kk) {
      int k0 = kk * 4 + kh;
      v2f a1, a2, b1, b2;
      a1.x = yr_[ti * 16 + M][k0];        a1.y = yr_[ti * 16 + M][k0 + 1];
      a2.x = s2 * yi_[ti * 16 + M][k0];   a2.y = s2 * yi_[ti * 16 + M][k0 + 1];
      b1.x = Bm1[k0][tj * 16 + M];        b1.y = Bm1[k0 + 1][tj * 16 + M];
      b2.x = Bm2[k0][tj * 16 + M];        b2.y = Bm2[k0 + 1][tj * 16 + M];
      acc = wmma4(a1, b1, acc);
      acc = wmma4(a2, b2, acc);
    }
    float (*D)[33] = part ? xi_ : xr_;
    #pragma unroll
    for (int v = 0; v < 8; ++v)
      D[ti * 16 + v + (lane >> 4) * 8][tj * 16 + M] = acc[v];
  }
  __syncthreads();

  // ---- store: X[k1 + 32*k2] = Z[k1][k2]  -> output element e has k1=e&31 ----
  if (flags & FLAG_SPLIT_OUT) {
    #pragma unroll
    for (int j = 0; j < 4; ++j) {
      int e = tid * 4 + j;
      float zr = xr_[e & 31][e >> 5];
      float zi = xi_[e & 31][e >> 5];
      if (flags & FLAG_CONJ_OUT) zi = -zi;
      size_t oi = base + (size_t)e * estride;
      outO[oi] = zr; outE[oi] = zi;
    }
  } else if (estride == 1) {
    #pragma unroll
    for (int j = 0; j < 4; ++j) {
      int e = tid * 4 + j;
      float zr = xr_[e & 31][e >> 5];
      float zi = xi_[e & 31][e >> 5];
      if (flags & FLAG_CONJ_OUT) zi = -zi;
      outC[base + (size_t)e] = make_float2(zr, zi);
    }
  } else {
#if USE_TDM
    #pragma unroll
    for (int j = 0; j < 4; ++j) {
      int e = tid * 4 + j;
      float zr = xr_[e & 31][e >> 5];
      float zi = xi_[e & 31][e >> 5];
      if (flags & FLAG_CONJ_OUT) zi = -zi;
      cbuf[e] = make_float2(zr, zi);
    }
    __syncthreads();
    if (tid < 32) {                       // wave 0 issues the DMA scatter
      u32x4 g0; i32x8 g1;
      tdm_desc((unsigned long long)(const void*)(outC + base),
               (unsigned)(size_t)(void*)cbuf, g0, g1);
      tdm_store(g0, g1);
    }
    // S_ENDPGM performs an implicit wait-idle (covers TENSORcnt).
#else
    #pragma unroll
    for (int j = 0; j < 4; ++j) {
      int e = tid * 4 + j;
      float zr = xr_[e & 31][e >> 5];
      float zi = xi_[e & 31][e >> 5];
      if (flags & FLAG_CONJ_OUT) zi = -zi;
      outC[base + (size_t)e * estride] = make_float2(zr, zi);
    }
#endif
  }
}

// ---------------------------------------------------------------------------
// filter(o,s) built on the fly in DFT layout, multiplied into F
// ---------------------------------------------------------------------------
__global__ __launch_bounds__(256)
void mul_filter_kernel(const float2* __restrict__ F, float2* __restrict__ out,
                       float invfo, float angl) {
  int idx = blockIdx.x * 256 + threadIdx.x;
  __builtin_prefetch(&F[idx], 0, 0);          // global_prefetch_b8
  int u = idx >> 10, v = idx & 1023;
  float fy = (float)((u < 512) ? u : u - 1024) * (1.0f / 1024.0f);
  float fx = (float)((v < 512) ? v : v - 1024) * (1.0f / 1024.0f);
  float filt = 0.0f;
  if (idx != 0) {
    float r = sqrtf(fx * fx + fy * fy);
    float lowpass = 1.0f / (1.0f + __expf(30.0f * __logf(r * (1.0f / 0.45f))));
    float lg = __logf(r * invfo);                       // log(radius/fo)
    float logGabor = __expf(-(lg * lg) * (1.0f / 0.71481813f)); // 2*ln(0.55)^2
    float theta = atan2f(-fy, fx);
    float d = theta - angl;
    float dtheta = fminf(fabsf(atan2f(__sinf(d), __cosf(d))) * 3.0f, PI_F);
    float spread = 0.5f * (__cosf(dtheta) + 1.0f);
    filt = spread * logGabor * lowpass;
  }
  float2 Fv = F[idx];
  out[idx] = make_float2(filt * Fv.x, filt * Fv.y);
}

// ---------------------------------------------------------------------------
// helpers for the median-based noise threshold tau / T
// ---------------------------------------------------------------------------
__global__ void zero_kernel(unsigned* p, int n) {
  int i = blockIdx.x * 256 + threadIdx.x;
  if (i < n) p[i] = 0u;
}

__global__ __launch_bounds__(256)
void an_max_kernel(const float* __restrict__ E0, const float* __restrict__ O0,
                   unsigned* __restrict__ mx) {
  __shared__ unsigned smax;
  int idx = blockIdx.x * 256 + threadIdx.x;
  if (threadIdx.x == 0) smax = 0u;
  __syncthreads();
  float an = sqrtf(E0[idx] * E0[idx] + O0[idx] * O0[idx]);
  atomicMax(&smax, __float_as_uint(an));      // an >= 0: uint order == float order
  __syncthreads();
  if (threadIdx.x == 0) atomicMax(mx, smax);
}

__global__ __launch_bounds__(256)
void hist_kernel(const float* __restrict__ E0, const float* __restrict__ O0,
                 const unsigned* __restrict__ mx, unsigned* __restrict__ hist) {
  __shared__ unsigned h[2048];
  for (int i = threadIdx.x; i < 2048; i += 256) h[i] = 0u;
  __syncthreads();
  int idx = blockIdx.x * 256 + threadIdx.x;
  float an = sqrtf(E0[idx] * E0[idx] + O0[idx] * O0[idx]);
  float mv = fmaxf(__uint_as_float(mx[0]), 1e-30f);
  int b = (int)(an * (2048.0f / mv));
  b = (b > 2047) ? 2047 : ((b < 0) ? 0 : b);
  atomicAdd(&h[b], 1u);
  __syncthreads();
  for (int i = threadIdx.x; i < 2048; i += 256)
    if (h[i]) atomicAdd(&hist[i], h[i]);
}

__global__ void tau_kernel(const unsigned* __restrict__ hist,
                           const unsigned* __restrict__ mx,
                           float* __restrict__ Tout) {
  float mv = fmaxf(__uint_as_float(mx[0]), 1e-30f);
  unsigned cum = 0; int b = 2047;
  for (int i = 0; i < 2048; ++i) {
    cum += hist[i];
    if (cum >= (unsigned)(NPIX / 2)) { b = i; break; }
  }
  float median = ((float)b + 0.5f) * mv * (1.0f / 2048.0f);
  float tau = median / sqrtf(logf(4.0f));
  float invm = 1.0f / 2.1f;
  float totalTau = tau * (1.0f - powf(invm, 5.0f)) / (1.0f - invm);
  Tout[0] = totalTau * (sqrtf(PI_F / 2.0f) + 2.0f * sqrtf((4.0f - PI_F) / 2.0f));
}

// ---------------------------------------------------------------------------
// per-orientation phase-congruency plane
// ---------------------------------------------------------------------------
__global__ __launch_bounds__(256)
void pc_kernel(const float* __restrict__ E, const float* __restrict__ O,
               const float* __restrict__ Tp, float* __restrict__ PC) {
  int idx = blockIdx.x * 256 + threadIdx.x;
  float e[5], o_[5], sE = 0.f, sO = 0.f, sA = 0.f, mA = 0.f;
  #pragma unroll
  for (int s = 0; s < 5; ++s) {
    e[s]  = E[(size_t)s * NPIX + idx];
    o_[s] = O[(size_t)s * NPIX + idx];
    float an = sqrtf(e[s] * e[s] + o_[s] * o_[s]);
    sE += e[s]; sO += o_[s]; sA += an; mA = fmaxf(mA, an);
  }
  float XE = sqrtf(sE * sE + sO * sO) + 1e-4f;
  float ME = sE / XE, MO = sO / XE;
  float En = 0.f;
  #pragma unroll
  for (int s = 0; s < 5; ++s)
    En += e[s] * ME + o_[s] * MO - fabsf(e[s] * MO - o_[s] * ME);
  En = fmaxf(En - Tp[0], 0.0f);
  float width  = (sA / (mA + 1e-4f) - 1.0f) * 0.25f;
  float weight = 1.0f / (1.0f + __expf((0.5f - width) * 10.0f));
  PC[idx] = weight * En / sA;
}

// ---------------------------------------------------------------------------
// moment analysis -> M, m
// ---------------------------------------------------------------------------
__global__ __launch_bounds__(256)
void final_kernel(const float* __restrict__ PCb, float* __restrict__ out) {
  int idx = blockIdx.x * 256 + threadIdx.x;
  float covx2 = 0.f, covy2 = 0.f, covxy = 0.f;
  #pragma unroll
  for (int o = 0; o < 6; ++o) {
    float pc = PCb[(size_t)o * NPIX + idx];
    float sn, cs;
    __sincosf((float)o * (PI_F / 6.0f), &sn, &cs);
    float cx = pc * cs, cy = pc * sn;
    covx2 += cx * cx; covy2 += cy * cy; covxy += cx * cy;
  }
  covx2 *= (1.0f / 3.0f); covy2 *= (1.0f / 3.0f); covxy *= (4.0f / 6.0f);
  float dd = covx2 - covy2;
  float denom = sqrtf(covxy * covxy + dd * dd) + 1e-4f;
  out[idx]        = (covy2 + covx2 + denom) * 0.5f;   // M
  out[NPIX + idx] = (covy2 + covx2 - denom) * 0.5f;   // m
}

// ---------------------------------------------------------------------------
extern "C" void kernel_launch(void* const* d_in, const int* in_sizes, int n_in,
                              void* d_out, int out_size, void* d_ws, size_t ws_size,
                              hipStream_t stream) {
  (void)in_sizes; (void)n_in; (void)out_size; (void)ws_size;
  const float* img = (const float*)d_in[0];
  float* out = (float*)d_out;
  float* ws  = (float*)d_ws;

  float2* F  = (float2*)(ws);
  float2* tA = (float2*)(ws + (size_t)2 * NPIX);
  float2* tB = (float2*)(ws + (size_t)4 * NPIX);
  float* Eb  = ws + (size_t)6  * NPIX;   // 30 planes
  float* Ob  = ws + (size_t)36 * NPIX;   // 30 planes
  float* PCb = ws + (size_t)66 * NPIX;   // 6 planes
  unsigned* mx   = (unsigned*)(ws + (size_t)72 * NPIX);
  unsigned* hist = mx + 1;               // 2048 bins
  float*    T    = (float*)(hist + 2048);

  // forward FFT2 of the image: rows then columns
  fft1024_kernel<<<1024, 256, 0, stream>>>(img, nullptr, tA, nullptr, nullptr,
                                           1, 1024, FLAG_REAL_IN);
  fft1024_kernel<<<1024, 256, 0, stream>>>(nullptr, tA, F, nullptr, nullptr,
                                           1024, 1, 0);

  for (int o = 0; o < 6; ++o) {
    float angl = (float)o * (float)M_PI / 6.0f;
    for (int s = 0; s < 5; ++s) {
      float invfo = 3.0f * powf(2.1f, (float)s);     // 1/fo_s
      mul_filter_kernel<<<4096, 256, 0, stream>>>(F, tA, invfo, angl);
      // unscaled ifft2 via conj trick: conj in on first pass, conj out on last
      fft1024_kernel<<<1024, 256, 0, stream>>>(nullptr, tA, tB, nullptr, nullptr,
                                               1024, 1, FLAG_CONJ_IN);
      float* Ep = Eb + (size_t)(o * 5 + s) * NPIX;
      float* Op = Ob + (size_t)(o * 5 + s) * NPIX;
      fft1024_kernel<<<1024, 256, 0, stream>>>(nullptr, tB, nullptr, Op, Ep,
                                               1, 1024, FLAG_CONJ_OUT | FLAG_SPLIT_OUT);
    }
    const float* E0 = Eb + (size_t)(o * 5) * NPIX;
    const float* O0 = Ob + (size_t)(o * 5) * NPIX;
    zero_kernel<<<9, 256, 0, stream>>>(mx, 1 + 2048);
    an_max_kernel<<<4096, 256, 0, stream>>>(E0, O0, mx);
    hist_kernel<<<4096, 256, 0, stream>>>(E0, O0, mx, hist);
    tau_kernel<<<1, 1, 0, stream>>>(hist, mx, T + o);
  }

  for (int o = 0; o < 6; ++o)
    pc_kernel<<<4096, 256, 0, stream>>>(Eb + (size_t)(o * 5) * NPIX,
                                        Ob + (size_t)(o * 5) * NPIX,
                                        T + o, PCb + (size_t)o * NPIX);

  final_kernel<<<4096, 256, 0, stream>>>(PCb, out);
}